// Net_25752623907081
// MI455X (gfx1250) — compile-verified
//
#include <hip/hip_runtime.h>

typedef float v2f __attribute__((ext_vector_type(2)));
typedef float v8f __attribute__((ext_vector_type(8)));

#define N_NODES 10000
#define N_EDGES 160000

// ---------------------------------------------------------------------------
// GEMM  Y[N, Fout] = X[N, Fin] @ W[Fin, Fout]   using V_WMMA_F32_16X16X4_F32.
// One wave computes FIVE 16x16 output tiles stacked in M (80 rows), so the
// strided B (weight) loads are amortized 5x:  per K-step of 4 we issue
// 5 contiguous v2f A-loads + 1 B pair for 5 WMMAs.
// (10000 rows = 125 blocks * 80 rows, exact.)
//
// A layout (16x4 f32): lane l: row m = l&15, K-pair base = 2*(l>>4) (contig)
// B layout (4x16 f32): lane l: col n = l&15, K-pair base = 2*(l>>4) (stride Fout)
// C/D (16x16 f32, v8f): element r: row = r + 8*(l>>4), col = l&15
// ---------------------------------------------------------------------------
__global__ __launch_bounds__(32) void gcn_gemm_wmma_f32(
    const float* __restrict__ X, const float* __restrict__ W,
    float* __restrict__ Y, int Fin, int Fout) {
  const int mt   = blockIdx.x;          // 0..124, each covers 80 rows
  const int nt   = blockIdx.y;          // Fout/16
  const int lane = threadIdx.x & 31;
  const int half = lane >> 4;
  const int l15  = lane & 15;

  const float* xrow = X + (size_t)(mt * 80 + l15) * Fin + 2 * half;
  const size_t xstep = (size_t)16 * Fin;
  const float* wcol  = W + (size_t)(2 * half) * Fout + (nt * 16 + l15);

  v8f acc0 = {}, acc1 = {}, acc2 = {}, acc3 = {}, acc4 = {};

#pragma unroll 2
  for (int k0 = 0; k0 < Fin; k0 += 4) {
    v2f b;
    b.x = wcol[(size_t)k0 * Fout];                // W[k0+2h  ][n]
    b.y = wcol[(size_t)(k0 + 1) * Fout];          // W[k0+2h+1][n]
    v2f a0 = *(const v2f*)(xrow + k0);
    v2f a1 = *(const v2f*)(xrow + xstep + k0);
    v2f a2 = *(const v2f*)(xrow + 2 * xstep + k0);
    v2f a3 = *(const v2f*)(xrow + 3 * xstep + k0);
    v2f a4 = *(const v2f*)(xrow + 4 * xstep + k0);
    acc0 = __builtin_amdgcn_wmma_f32_16x16x4_f32(false, a0, false, b, (short)0,
                                                 acc0, false, false);
    acc1 = __builtin_amdgcn_wmma_f32_16x16x4_f32(false, a1, false, b, (short)0,
                                                 acc1, false, false);
    acc2 = __builtin_amdgcn_wmma_f32_16x16x4_f32(false, a2, false, b, (short)0,
                                                 acc2, false, false);
    acc3 = __builtin_amdgcn_wmma_f32_16x16x4_f32(false, a3, false, b, (short)0,
                                                 acc3, false, false);
    acc4 = __builtin_amdgcn_wmma_f32_16x16x4_f32(false, a4, false, b, (short)0,
                                                 acc4, false, false);
  }

  float* yp = Y + (size_t)(mt * 80 + half * 8) * Fout + (nt * 16 + l15);
  const size_t ystep = (size_t)16 * Fout;
#pragma unroll
  for (int r = 0; r < 8; ++r) yp[(size_t)r * Fout] = acc0[r];
  yp += ystep;
#pragma unroll
  for (int r = 0; r < 8; ++r) yp[(size_t)r * Fout] = acc1[r];
  yp += ystep;
#pragma unroll
  for (int r = 0; r < 8; ++r) yp[(size_t)r * Fout] = acc2[r];
  yp += ystep;
#pragma unroll
  for (int r = 0; r < 8; ++r) yp[(size_t)r * Fout] = acc3[r];
  yp += ystep;
#pragma unroll
  for (int r = 0; r < 8; ++r) yp[(size_t)r * Fout] = acc4[r];
}

// ---------------------------------------------------------------------------
// Degree count (float so the mean divide is direct). Computed once per launch.
// ---------------------------------------------------------------------------
__global__ void gcn_degrees(const int* __restrict__ dst, float* __restrict__ deg,
                            int e_count) {
  int e = blockIdx.x * blockDim.x + threadIdx.x;
  if (e < e_count) atomicAdd(&deg[dst[e]], 1.0f);
}

// ---------------------------------------------------------------------------
// Edge scatter: Agg[dst[e], :] += Y[src[e], :], float4 per thread.
// qshift = log2(F) - 2.  Consecutive threads cover consecutive quads of one
// edge -> coalesced 128B loads and coalesced atomics.
// ---------------------------------------------------------------------------
__global__ void gcn_scatter(const float* __restrict__ Y,
                            const int* __restrict__ src,
                            const int* __restrict__ dst,
                            float* __restrict__ Agg, int qshift, int total4) {
  int idx = blockIdx.x * blockDim.x + threadIdx.x;
  if (idx >= total4) return;
  int e = idx >> qshift;
  int q = idx & ((1 << qshift) - 1);
  int s = src[e];
  int d = dst[e];
  const float4 v = ((const float4*)(Y + (((size_t)s) << (qshift + 2))))[q];
  float* ap = Agg + (((size_t)d) << (qshift + 2)) + 4 * q;
  atomicAdd(ap + 0, v.x);
  atomicAdd(ap + 1, v.y);
  atomicAdd(ap + 2, v.z);
  atomicAdd(ap + 3, v.w);
}

// ---------------------------------------------------------------------------
// Fused mean / isolated-node select / bias / ReLU  (float4 per thread).
// ---------------------------------------------------------------------------
__global__ void gcn_finalize_relu(const float* __restrict__ Y,
                                  const float* __restrict__ Agg,
                                  const float* __restrict__ deg,
                                  const float* __restrict__ bias,
                                  float* __restrict__ Xout, int qshift,
                                  int total4) {
  int idx = blockIdx.x * blockDim.x + threadIdx.x;
  if (idx >= total4) return;
  int n = idx >> qshift;
  int q = idx & ((1 << qshift) - 1);
  float dg = deg[n];
  float4 a = ((const float4*)Agg)[idx];
  float4 y = ((const float4*)Y)[idx];
  float4 b = ((const float4*)bias)[q];
  float inv = (dg > 0.0f) ? (1.0f / dg) : 0.0f;
  float4 r;
  r.x = (dg > 0.0f ? a.x * inv : y.x) + b.x;
  r.y = (dg > 0.0f ? a.y * inv : y.y) + b.y;
  r.z = (dg > 0.0f ? a.z * inv : y.z) + b.z;
  r.w = (dg > 0.0f ? a.w * inv : y.w) + b.w;
  r.x = r.x > 0.0f ? r.x : 0.0f;
  r.y = r.y > 0.0f ? r.y : 0.0f;
  r.z = r.z > 0.0f ? r.z : 0.0f;
  r.w = r.w > 0.0f ? r.w : 0.0f;
  ((float4*)Xout)[idx] = r;
}

// ---------------------------------------------------------------------------
// Final layer: mean/select/bias then softmax over F=256 features (one block
// of 256 threads per node; tree reductions in LDS).
// ---------------------------------------------------------------------------
__global__ __launch_bounds__(256) void gcn_finalize_softmax(
    const float* __restrict__ Y, const float* __restrict__ Agg,
    const float* __restrict__ deg, const float* __restrict__ bias,
    float* __restrict__ out) {
  const int n = blockIdx.x;
  const int f = threadIdx.x;          // 0..255
  const size_t idx = (size_t)n * 256 + f;

  float dg = deg[n];
  float v  = ((dg > 0.0f) ? (Agg[idx] / dg) : Y[idx]) + bias[f];

  __shared__ float red[256];
  red[f] = v;
  __syncthreads();
#pragma unroll
  for (int s = 128; s > 0; s >>= 1) {
    if (f < s) red[f] = fmaxf(red[f], red[f + s]);
    __syncthreads();
  }
  float mx = red[0];
  __syncthreads();

  float e = __expf(v - mx);
  red[f] = e;
  __syncthreads();
#pragma unroll
  for (int s = 128; s > 0; s >>= 1) {
    if (f < s) red[f] += red[f + s];
    __syncthreads();
  }
  out[idx] = e / red[0];
}

// ---------------------------------------------------------------------------
// Host-side orchestration.
// Inputs: 0=features 1=src 2=dst 3=W1 4=b1 5=W2 6=b2 7=W3 8=b3 9=W4 10=b4
// ---------------------------------------------------------------------------
extern "C" void kernel_launch(void* const* d_in, const int* in_sizes, int n_in,
                              void* d_out, int out_size, void* d_ws,
                              size_t ws_size, hipStream_t stream) {
  const float* feat = (const float*)d_in[0];
  const int*   src  = (const int*)d_in[1];
  const int*   dst  = (const int*)d_in[2];
  const float* W[4] = {(const float*)d_in[3], (const float*)d_in[5],
                       (const float*)d_in[7], (const float*)d_in[9]};
  const float* B[4] = {(const float*)d_in[4], (const float*)d_in[6],
                       (const float*)d_in[8], (const float*)d_in[10]};

  // Workspace layout (floats): deg | Y | Agg | Xa | Xb   (big bufs 10000x512)
  const size_t BUF = (size_t)N_NODES * 512;
  float* ws   = (float*)d_ws;
  float* deg  = ws;
  float* Ybuf = ws + 16384;
  float* Agg  = Ybuf + BUF;
  float* Xa   = Agg + BUF;
  float* Xb   = Xa + BUF;

  // Degrees (invariant across layers; recomputed each launch -> deterministic)
  hipMemsetAsync(deg, 0, N_NODES * sizeof(float), stream);
  gcn_degrees<<<(N_EDGES + 255) / 256, 256, 0, stream>>>(dst, deg, N_EDGES);

  const int   Fin[4]    = {2048, 128, 256, 512};
  const int   Fout[4]   = {128, 256, 512, 256};
  const int   fshift[4] = {7, 8, 9, 8};
  const float* Xin[4]   = {feat, Xa, Xb, Xa};
  float*       Xout[3]  = {Xa, Xb, Xa};

  for (int l = 0; l < 4; ++l) {
    const int fo = Fout[l];
    const int qshift = fshift[l] - 2;
    // 1) projection GEMM on the WMMA path (5 M-tiles per wave)
    gcn_gemm_wmma_f32<<<dim3(N_NODES / 80, fo / 16), 32, 0, stream>>>(
        Xin[l], W[l], Ybuf, Fin[l], fo);
    // 2) aggregate projected features over edges (float4 granularity)
    hipMemsetAsync(Agg, 0, (size_t)N_NODES * fo * sizeof(float), stream);
    int etotal4 = N_EDGES * (fo / 4);
    gcn_scatter<<<(etotal4 + 255) / 256, 256, 0, stream>>>(Ybuf, src, dst, Agg,
                                                           qshift, etotal4);
    // 3) mean / select / bias / activation
    if (l < 3) {
      int ntotal4 = N_NODES * (fo / 4);
      gcn_finalize_relu<<<(ntotal4 + 255) / 256, 256, 0, stream>>>(
          Ybuf, Agg, deg, B[l], Xout[l], qshift, ntotal4);
    } else {
      gcn_finalize_softmax<<<N_NODES, 256, 0, stream>>>(Ybuf, Agg, deg, B[l],
                                                        (float*)d_out);
    }
  }
}